// SmallSMBlock3D_45492293599241
// MI455X (gfx1250) — compile-verified
//
#include <hip/hip_runtime.h>
#include <stdint.h>

// Fused SmallSMBlock3D for MI455X (gfx1250, wave32):
//  stage 1: K = conv3d(image) as im2col GEMM on V_WMMA_F32_16X16X4_F32
//  stage 2: per-voxel 3x3x3 dynamic stencil on VALU, K cached in VGPRs,
//           reused across 16 batches
//  data movement: TDM 3D tile DMA (tensor_load_to_lds), double-buffered and
//  pipelined on TENSORcnt (issue tile bi+1, wait tensorcnt<=1 for tile bi)

#define TD 4
#define THH 4
#define TW 16
#define NV 256                 // voxels per block
#define HD 6
#define HHH 6
#define HW 18
#define HN (HD * HHH * HW)     // 648 halo elements
#define VOX (128 * 128 * 128)

typedef __attribute__((ext_vector_type(2))) float v2f;
typedef __attribute__((ext_vector_type(8))) float v8f;
typedef __attribute__((ext_vector_type(4))) unsigned int u32x4;
typedef __attribute__((ext_vector_type(8))) int i32x8;
typedef __attribute__((ext_vector_type(4))) int i32x4;

#if __has_builtin(__builtin_amdgcn_tensor_load_to_lds) && __has_builtin(__builtin_amdgcn_s_wait_tensorcnt)
#define HAVE_TDM 1
#if __has_include(<hip/amd_detail/amd_gfx1250_TDM.h>)
#define TDM_SIX 1
#endif
#endif

#if __has_builtin(__builtin_amdgcn_wmma_f32_16x16x4_f32)
#define HAVE_WMMA 1
#endif

#if defined(HAVE_TDM)
static constexpr bool kUseTdm = true;
#else
static constexpr bool kUseTdm = false;
#endif

// tap index (0..26) -> element offset inside the 6x6x18 halo tile.
// constexpr so every use below folds to an inline constant after unrolling.
constexpr int tap_off_c(int k) {
  return (k / 9) * (HHH * HW) + ((k % 9) / 3) * HW + ((k % 9) % 3);
}

// Issue (no wait) a 3D tile DMA: 6x6x18 f32 tile, strides 128 / 128*128,
// global -> LDS. D# packed per CDNA5 ISA.
// Issued by wave 0 only (TDM ignores EXEC; one op moves the whole tile).
__device__ __forceinline__ void tdm_tile_issue(const float* gtile, float* ldsDst) {
#if defined(HAVE_TDM)
  if (threadIdx.x < 32) {
    unsigned long long ga = (unsigned long long)(uintptr_t)gtile;
    unsigned ldsOff = (unsigned)(uintptr_t)ldsDst;   // low 32 bits of generic = LDS byte addr
    u32x4 g0 = {0u, 0u, 0u, 0u};
    g0[0] = 1u;                                        // count=1 (valid), user mode, no gather
    g0[1] = ldsOff;                                    // lds_addr (bytes)
    g0[2] = (unsigned)(ga & 0xffffffffull);            // global_addr[31:0]
    g0[3] = (unsigned)((ga >> 32) & 0x1ffffffull) | (2u << 30);  // ga[56:32], type=2
    i32x8 g1 = {0, 0, 0, 0, 0, 0, 0, 0};
    g1[0] = (int)(2u << 16);                           // wg_mask=0, data_size=4B
    g1[1] = (int)(128u << 16);                         // tensor_dim0 = 128
    g1[2] = (int)(128u << 16);                         // dim0 hi=0; tensor_dim1 = 128
    g1[3] = (int)((unsigned)HW << 16);                 // dim1 hi=0; tile_dim0 = 18
    g1[4] = (int)((unsigned)HHH | ((unsigned)HD << 16)); // tile_dim1=6, tile_dim2=6
    g1[5] = 128;                                       // tensor_dim0_stride = 128 elems
    g1[6] = (int)((128u * 128u) << 16);                // stride0 hi=0; tensor_dim1_stride lo16
    g1[7] = (int)((128u * 128u) >> 16);                // tensor_dim1_stride hi = 0
    i32x4 g2 = {0, 0, 0, 0};
    g2[0] = 128;                                       // tensor_dim2
    g2[1] = 0;                                         // tensor_dim3 unused
    g2[2] = 128 * 128 * 128;                           // tensor_dim2_stride
    g2[3] = 0;                                         // stride2 hi, tile_dim3=0
    i32x4 g3 = {0, 0, 0, 0};
#if defined(TDM_SIX)
    i32x8 g4 = {0, 0, 0, 0, 0, 0, 0, 0};
    __builtin_amdgcn_tensor_load_to_lds(g0, g1, g2, g3, g4, 0);
#else
    __builtin_amdgcn_tensor_load_to_lds(g0, g1, g2, g3, 0);
#endif
  }
#else
  (void)gtile; (void)ldsDst;
#endif
}

__device__ __forceinline__ void tdm_wait0() {
#if defined(HAVE_TDM)
  if (threadIdx.x < 32) __builtin_amdgcn_s_wait_tensorcnt((short)0);
#endif
}
__device__ __forceinline__ void tdm_wait1() {
#if defined(HAVE_TDM)
  if (threadIdx.x < 32) __builtin_amdgcn_s_wait_tensorcnt((short)1);
#endif
}

// Boundary fallback: predicated scalar gather with zero padding.
__device__ __forceinline__ void scalar_tile_load(const float* gbase, int d0, int h0, int w0,
                                                 float* ldsDst) {
  for (int i = threadIdx.x; i < HN; i += 256) {
    int z = i / (HHH * HW);
    int r = i - z * (HHH * HW);
    int y = r / HW;
    int xx = r - y * HW;
    int gz = d0 - 1 + z, gy = h0 - 1 + y, gx = w0 - 1 + xx;
    float v = 0.f;
    if ((unsigned)gz < 128u && (unsigned)gy < 128u && (unsigned)gx < 128u)
      v = gbase[((size_t)gz * 128 + gy) * 128 + gx];
    ldsDst[i] = v;
  }
}

__global__ __launch_bounds__(256)
void smallsm3d_fused(const float* __restrict__ img, const float* __restrict__ xg,
                     const float* __restrict__ Wg, const float* __restrict__ bg,
                     float* __restrict__ out) {
  __shared__ float sImg[HN];        // image halo tile
  __shared__ float sX[2][HN];       // x halo tiles, double buffered
  __shared__ float sK[27 * NV];     // per-voxel 27-tap kernel for this tile
  __shared__ float sW[32 * 28];     // stage-1 weights, zero padded to 32x28
  __shared__ float sB[32];

  const int w0 = blockIdx.x * TW;
  const int h0 = blockIdx.y * THH;
  const int d0 = blockIdx.z * TD;
  const bool interior = (d0 >= 1) && (d0 + TD <= 127) &&
                        (h0 >= 1) && (h0 + THH <= 127) &&
                        (w0 >= 1) && (w0 + TW <= 127);
  const size_t tileOff = ((size_t)(d0 - 1) * 128 + (h0 - 1)) * 128 + (w0 - 1);

  // Stage-1 weights + bias -> LDS (zero padded so WMMA K/M padding is free).
  for (int i = threadIdx.x; i < 32 * 28; i += 256) {
    int c = i / 28, k = i - c * 28;
    sW[i] = (c < 27 && k < 27) ? Wg[c * 27 + k] : 0.f;
  }
  if (threadIdx.x < 32) sB[threadIdx.x] = (threadIdx.x < 27) ? bg[threadIdx.x] : 0.f;

  // Image tile + first x tile: issue both DMAs, wait only for the image
  // (x batch-0 DMA continues under the stage-1 WMMA GEMM).
  if (kUseTdm && interior) {
    tdm_tile_issue(img + tileOff, sImg);
    tdm_tile_issue(xg + tileOff, sX[0]);
    tdm_wait1();                       // image complete; x0 may stay in flight
  } else {
    scalar_tile_load(img, d0, h0, w0, sImg);
    scalar_tile_load(xg, d0, h0, w0, sX[0]);
  }
  __syncthreads();

  // ---- Stage 1: K[27 x 256] = W[27x27] x P[27x256] + b, via f32 WMMA 16x16x4 ----
#if defined(HAVE_WMMA)
  {
    const int lid = threadIdx.x & 31;
    const int wid = threadIdx.x >> 5;     // wave 0..7
    const int lane15 = lid & 15;
    const int khalf = lid >> 4;           // lanes 16-31 carry K+2 (f32 A/B layout)

    // A fragments: loaded once, reused for both N-tiles.
    v2f afrag[2][7];
    #pragma unroll
    for (int mt = 0; mt < 2; ++mt) {
      const int c = mt * 16 + lane15;
      #pragma unroll
      for (int s = 0; s < 7; ++s) {
        const int k0 = 4 * s + 2 * khalf;   // zero-padded rows/cols in sW
        afrag[mt][s][0] = sW[c * 28 + k0];
        afrag[mt][s][1] = sW[c * 28 + k0 + 1];
      }
    }

    #pragma unroll
    for (int nn = 0; nn < 2; ++nn) {
      const int nt = wid * 2 + nn;        // 16 N-tiles over 8 waves
      const int n2 = nt * 16 + lane15;    // voxel column for this lane
      const int vz = n2 >> 6, vy = (n2 >> 4) & 3, vx = n2 & 15;
      const int ibase = vz * (HHH * HW) + vy * HW + vx;

      // B fragments: tap offsets fold to inline constants (cndmask on khalf),
      // no runtime div-by-9/3 chains. Shared by both M-tiles.
      v2f bfrag[7];
      #pragma unroll
      for (int s = 0; s < 7; ++s) {
        const int o0 = khalf ? tap_off_c(4 * s + 2) : tap_off_c(4 * s);
        const int o1 = khalf ? tap_off_c(4 * s + 3) : tap_off_c(4 * s + 1);
        bfrag[s][0] = sImg[ibase + o0];                  // k <= 26: always valid
        float v1 = sImg[ibase + o1];                     // k=27 slot zeroed below
        bfrag[s][1] = (s == 6 && khalf) ? 0.f : v1;
      }

      #pragma unroll
      for (int mt = 0; mt < 2; ++mt) {    // channels 0-15, 16-26(+pad)
        v8f acc = {0.f, 0.f, 0.f, 0.f, 0.f, 0.f, 0.f, 0.f};
        #pragma unroll
        for (int s = 0; s < 7; ++s)
          acc = __builtin_amdgcn_wmma_f32_16x16x4_f32(
              false, afrag[mt][s], false, bfrag[s], (short)0, acc, false, false);
        #pragma unroll
        for (int r = 0; r < 8; ++r) {     // D layout: VGPR r = row r (+8 for hi lanes)
          const int c = mt * 16 + khalf * 8 + r;
          if (c < 27) sK[c * NV + n2] = acc[r] + sB[c];
        }
      }
    }
  }
#else
  // VALU fallback (also what the host pass parses)
  for (int i = threadIdx.x; i < 27 * NV; i += 256) {
    int c = i >> 8, nn = i & 255;
    int vz = nn >> 6, vy = (nn >> 4) & 3, vx = nn & 15;
    int ibase = vz * (HHH * HW) + vy * HW + vx;
    float acc = sB[c];
    #pragma unroll
    for (int t = 0; t < 27; ++t)
      acc += sW[c * 28 + t] * sImg[ibase + tap_off_c(t)];
    sK[i] = acc;
  }
#endif
  __syncthreads();

  // ---- Stage 2: per-voxel dynamic 3x3x3 stencil, K cached in VGPRs ----
  const int n = threadIdx.x;
  const int vz = n >> 6, vy = (n >> 4) & 3, vx = n & 15;
  const int gz = d0 + vz, gy = h0 + vy, gx = w0 + vx;

  float kreg[27];
  #pragma unroll
  for (int t = 0; t < 27; ++t) kreg[t] = sK[t * NV + n];

  for (int bi = 0; bi < 16; ++bi) {
    const int cur = bi & 1;
    // Issue next batch's tile into the alternate buffer (DMA overlaps compute).
    if (bi + 1 < 16) {
      const float* xb1 = xg + (size_t)(bi + 1) * VOX;
      if (kUseTdm && interior) tdm_tile_issue(xb1 + tileOff, sX[cur ^ 1]);
      else                     scalar_tile_load(xb1, d0, h0, w0, sX[cur ^ 1]);
    }
    // In-order TENSORcnt: <=1 outstanding means tile bi has landed.
    if (kUseTdm && interior) {
      if (bi + 1 < 16) tdm_wait1();
      else             tdm_wait0();
    }
    // L2 prefetch two batches ahead.
    if (bi + 2 < 16 && threadIdx.x < HD * HHH) {
      int z = threadIdx.x / HHH, y = threadIdx.x - z * HHH;
      int pz = d0 - 1 + z; pz = pz < 0 ? 0 : (pz > 127 ? 127 : pz);
      int py = h0 - 1 + y; py = py < 0 ? 0 : (py > 127 ? 127 : py);
      int px = w0 - 1;     px = px < 0 ? 0 : px;
      __builtin_prefetch(xg + (size_t)(bi + 2) * VOX + ((size_t)pz * 128 + py) * 128 + px, 0, 0);
    }
    __syncthreads();

    const float* xs = sX[cur];
    float acc = 0.f;
    #pragma unroll
    for (int t = 0; t < 27; ++t)
      acc += kreg[t] * xs[(vz + (t / 9)) * (HHH * HW) +
                          (vy + ((t % 9) / 3)) * HW + (vx + (t % 3))];
    out[(((size_t)bi * 128 + gz) * 128 + gy) * 128 + gx] = acc;
    __syncthreads();
  }
}

extern "C" void kernel_launch(void* const* d_in, const int* in_sizes, int n_in,
                              void* d_out, int out_size, void* d_ws, size_t ws_size,
                              hipStream_t stream) {
  (void)in_sizes; (void)n_in; (void)out_size; (void)d_ws; (void)ws_size;
  const float* img = (const float*)d_in[0];   // [1,128,128,128]
  const float* x   = (const float*)d_in[1];   // [16,1,128,128,128]
  const float* W   = (const float*)d_in[2];   // [27,1,3,3,3]
  const float* b   = (const float*)d_in[3];   // [27]
  float* out = (float*)d_out;                 // [16,1,128,128,128]
  dim3 grid(128 / TW, 128 / THH, 128 / TD);   // (8, 32, 32)
  smallsm3d_fused<<<grid, dim3(256), 0, stream>>>(img, x, W, b, out);
}